// SpikeCountLayer_83150566851382
// MI455X (gfx1250) — compile-verified
//
#include <hip/hip_runtime.h>
#include <stdint.h>

// ---------------------------------------------------------------------------
// SpikeCountLayer histogram for MI455X (gfx1250).
//   input : int32 [B=32, T=1024, H=32, W=32], spike ids in [0,32)
//   output: int32 [B, 32, H, W], out[b,s,p] = #{t : in[b,t,p]==s}
// HBM-bound streaming histogram: async global->LDS staging (ASYNCcnt path),
// thread-private LDS bins via ds_add_u32, deterministic integer global
// atomics to merge T-chunk partials.
// ---------------------------------------------------------------------------

static constexpr int kB             = 32;
static constexpr int kT             = 1024;
static constexpr int kHW            = 1024;                 // 32*32 pixels (contiguous per (b,t))
static constexpr int kS             = 32;                   // bins
static constexpr int kChunks        = 16;                   // T-chunks per batch -> 512 WGs
static constexpr int kTChunk        = kT / kChunks;         // 64 planes per WG
static constexpr int kThreads       = 256;                  // 8 waves (wave32)
static constexpr int kPlaneBytes    = kHW * 4;              // 4096 B per time-plane
static constexpr int kPlanesPerStage= 4;
static constexpr int kStages        = kTChunk / kPlanesPerStage;  // 16
static constexpr int kStageBytes    = kPlanesPerStage * kPlaneBytes; // 16 KB
static constexpr int kHistWords     = kHW * kS;             // 32768
static constexpr int kHistBytes     = kHistWords * 4;       // 128 KB
static constexpr int kLdsBytes      = kHistBytes + 2 * kStageBytes; // 160 KB (2 WGs / 320KB WGP)

// ---- CDNA5 async-counter wait -------------------------------------------------
#if defined(__has_builtin)
#if __has_builtin(__builtin_amdgcn_s_wait_asynccnt)
#define WAIT_ASYNCCNT(n) __builtin_amdgcn_s_wait_asynccnt(n)
#endif
#endif
#ifndef WAIT_ASYNCCNT
#define WAIT_ASYNCCNT(n) asm volatile("s_wait_asynccnt %0" ::"n"(n) : "memory")
#endif

// ---- CDNA5 async global->LDS load (GVS mode) ---------------------------------
// global_load_async_to_lds_b128 vdst, vaddr, saddr offset:IMM
//   copies 16 bytes per lane from the global address saddr plus vaddr to the LDS address in vdst (ASYNCcnt)
// Note: IMM offsets BOTH the LDS and the global address (CDNA5 ISA),
// which we exploit to address the 4 planes of a stage with one vdst/voff pair.
template <int IMM>
__device__ __forceinline__ void async_load_b128(uint32_t lds_byte_addr, uint32_t voff,
                                                uint64_t sbase) {
  asm volatile("global_load_async_to_lds_b128 %0, %1, %2 offset:%3"
               :: "v"(lds_byte_addr), "v"(voff), "s"(sbase), "n"(IMM)
               : "memory");
}

__device__ __forceinline__ void issue_stage(int g, uint32_t stage_lds0, uint32_t my16,
                                            uint64_t gbase) {
  // double buffer: stage g lives in buffer (g&1)
  uint32_t vdst = stage_lds0 + ((g & 1) ? (uint32_t)kStageBytes : 0u) + my16;
  uint32_t voff = (uint32_t)g * (uint32_t)kStageBytes + my16;
  async_load_b128<0 * kPlaneBytes>(vdst, voff, gbase);
  async_load_b128<1 * kPlaneBytes>(vdst, voff, gbase);
  async_load_b128<2 * kPlaneBytes>(vdst, voff, gbase);
  async_load_b128<3 * kPlaneBytes>(vdst, voff, gbase);
}

extern __shared__ uint32_t smem[];

__global__ void __launch_bounds__(kThreads)
spike_hist_kernel(const int* __restrict__ in, int* __restrict__ out) {
  const int tid   = (int)threadIdx.x;
  const int b     = (int)blockIdx.x / kChunks;
  const int chunk = (int)blockIdx.x % kChunks;

  uint32_t* hist = smem;                      // [pixel(1024)][bin(32)] u32, bank-rotated
  const uint32_t* stage = smem + kHistWords;  // 2 x 16KB staging buffers

  // LDS byte offset of dynamic-shared base (truncate the aperture address).
  const uint32_t lds0       = (uint32_t)(uintptr_t)(void*)smem;
  const uint32_t stage_lds0 = lds0 + (uint32_t)kHistBytes;
  const uint32_t my16       = (uint32_t)tid * 16u;
  const uint32_t p0         = (uint32_t)tid * 4u;   // first of this thread's 4 pixels

  // Global base of this WG's 64 contiguous time-planes.
  const uint64_t gbase =
      (uint64_t)(uintptr_t)(in + ((size_t)b * kT + (size_t)chunk * kTChunk) * (size_t)kHW);

  // Zero exactly this thread's own bins (pixels p0..p0+3 -> words p*32..p*32+31).
  {
    uint4 z = make_uint4(0u, 0u, 0u, 0u);
    uint4* h4 = (uint4*)hist;
#pragma unroll
    for (int j = 0; j < 32; ++j) h4[tid * 32 + j] = z;
  }
  // No barrier needed: bins are thread-private until the epilogue.

  issue_stage(0, stage_lds0, my16, gbase);

  for (int g = 0; g < kStages; ++g) {
    if (g + 1 < kStages) {
      issue_stage(g + 1, stage_lds0, my16, gbase);
      WAIT_ASYNCCNT(4);   // stage g's 4 loads done; stage g+1 still in flight
    } else {
      WAIT_ASYNCCNT(0);
    }
    const uint32_t* buf = stage + ((g & 1) ? (kStageBytes / 4) : 0);
#pragma unroll
    for (int k = 0; k < kPlanesPerStage; ++k) {
      // Each thread consumes exactly the 16 bytes it staged: no cross-lane hazard.
      uint4 v = *(const uint4*)(buf + k * kHW + p0);
      // Thread-private bins; ds_add_u32 (no return) is 1 LDS op/element.
      // Rotation (s+p)&31 decorrelates data-dependent bank conflicts.
      atomicAdd(&hist[(p0 + 0u) * kS + ((v.x + p0 + 0u) & (kS - 1))], 1u);
      atomicAdd(&hist[(p0 + 1u) * kS + ((v.y + p0 + 1u) & (kS - 1))], 1u);
      atomicAdd(&hist[(p0 + 2u) * kS + ((v.z + p0 + 2u) & (kS - 1))], 1u);
      atomicAdd(&hist[(p0 + 3u) * kS + ((v.w + p0 + 3u) & (kS - 1))], 1u);
    }
  }

  __syncthreads();  // epilogue reads other threads' bins

  // Merge partials: out[(b*32+s)*1024 + p] += hist[p][rot(s,p)].
  // Consecutive lanes -> consecutive p (coalesced, conflict-free LDS reads).
  const int out_base = b * (kS * kHW);
  for (int idx = tid; idx < kS * kHW; idx += kThreads) {
    int s = idx >> 10;          // idx = s*1024 + p
    int p = idx & (kHW - 1);
    uint32_t c = hist[(uint32_t)p * kS + (((uint32_t)s + (uint32_t)p) & (kS - 1))];
    atomicAdd((unsigned int*)(out + out_base + idx), c);  // integer: deterministic
  }
}

__global__ void zero_out_kernel(int* __restrict__ out, int n) {
  int i  = (int)(blockIdx.x * blockDim.x + threadIdx.x);
  int n4 = n >> 2;
  if (i < n4) ((int4*)out)[i] = make_int4(0, 0, 0, 0);
  int tail = n & 3;
  if (i < tail) out[n4 * 4 + i] = 0;
}

extern "C" void kernel_launch(void* const* d_in, const int* in_sizes, int n_in,
                              void* d_out, int out_size, void* d_ws, size_t ws_size,
                              hipStream_t stream) {
  (void)in_sizes; (void)n_in; (void)d_ws; (void)ws_size;
  const int* in = (const int*)d_in[0];   // d_in[1] = dim_s (==32, hardcoded)
  int* out = (int*)d_out;                // int32 output (4B elements)

  int n4 = (out_size + 3) / 4;
  zero_out_kernel<<<(n4 + 255) / 256, 256, 0, stream>>>(out, out_size);
  spike_hist_kernel<<<kB * kChunks, kThreads, kLdsBytes, stream>>>(in, out);
}